// SampledSoftmaxCrossEntropy_55293408968971
// MI455X (gfx1250) — compile-verified
//
#include <hip/hip_runtime.h>
#include <math.h>

// Sampled softmax cross entropy for MI455X (gfx1250, wave32).
// B=32768 rows, D=512, S=100 negatives, out = [logits(B x 101), loss].

#define B_ROWS 32768
#define DIM    512
#define NSAMP  100
#define SPAD   112          // NSAMP padded to 7 tiles of 16
#define NTILES 7
#define COLS   101          // output row stride (true logit + 100 sampled)
#define KC     32           // K chunk staged in LDS (double buffered)
#define LDT    (KC + 4)     // padded LDS row stride: 144B rows -> 16B-aligned
                            // b128 stores, 36 mod 64 banks -> conflict-free frags

typedef float v2f __attribute__((ext_vector_type(2)));
typedef float v8f __attribute__((ext_vector_type(8)));

// Async global->LDS copy of 16 bytes (tracked by ASYNCcnt, no VGPR round trip).
__device__ __forceinline__ void async_ld128(uint32_t lds_off, const float* gaddr) {
  asm volatile("global_load_async_to_lds_b128 %0, %1, off"
               :: "v"(lds_off), "v"((unsigned long long)(uintptr_t)gaddr)
               : "memory");
}
__device__ __forceinline__ void wait_async0() {
  asm volatile("s_wait_asynccnt 0x0" ::: "memory");
}

// ---------------------------------------------------------------------------
// 1) Gather sampled weight rows into a compact, zero-padded [SPAD][DIM]
//    buffer and precompute per-column bias - log(q).
// ---------------------------------------------------------------------------
__global__ void gather_kernel(const float* __restrict__ weights,
                              const float* __restrict__ biases,
                              const int*   __restrict__ sampled_ids,
                              const float* __restrict__ sexp,
                              float* __restrict__ sw,
                              float* __restrict__ colbias) {
  int tid = blockIdx.x * blockDim.x + threadIdx.x;
  if (tid < SPAD) {
    float cb = 0.f;
    if (tid < NSAMP) cb = biases[sampled_ids[tid]] - logf(sexp[tid]);
    colbias[tid] = cb;
  }
  int stride = gridDim.x * blockDim.x;
  for (int i = tid; i < SPAD * DIM; i += stride) {
    int s = i >> 9;             // / DIM
    int d = i & (DIM - 1);
    float v = 0.f;
    if (s < NSAMP) v = weights[(size_t)sampled_ids[s] * DIM + d];
    sw[i] = v;
  }
}

// ---------------------------------------------------------------------------
// 2) True logits: one wave32 per row, coalesced float4 loads, shfl reduce.
//    out[b*COLS + 0] = dot(inputs[b], weights[label[b]]) + bias - log(q_b)
// ---------------------------------------------------------------------------
__global__ void true_logits_kernel(const float* __restrict__ inputs,
                                   const float* __restrict__ weights,
                                   const float* __restrict__ biases,
                                   const int*   __restrict__ label,
                                   const float* __restrict__ texp,
                                   float* __restrict__ out) {
  int wid  = (blockIdx.x * blockDim.x + threadIdx.x) >> 5;  // one wave per row
  int lane = threadIdx.x & 31;
  if (wid >= B_ROWS) return;
  int lbl = label[wid];
  const float4* xi = (const float4*)(inputs  + (size_t)wid * DIM);
  const float4* wv = (const float4*)(weights + (size_t)lbl * DIM);
  float s = 0.f;
#pragma unroll
  for (int i = 0; i < (DIM / 4) / 32; ++i) {   // 4 float4 per lane
    float4 a = xi[lane + 32 * i];
    float4 b = wv[lane + 32 * i];
    s += a.x * b.x + a.y * b.y + a.z * b.z + a.w * b.w;
  }
#pragma unroll
  for (int off = 16; off > 0; off >>= 1) s += __shfl_xor(s, off, 32);
  if (lane == 0)
    out[(size_t)wid * COLS] = s + biases[lbl] - logf(texp[wid]);
}

// ---------------------------------------------------------------------------
// 3) Sampled logits GEMM via V_WMMA_F32_16X16X4_F32 (exact fp32).
//    M=32768, N=112 (7 tiles, last 12 cols masked), K=512.
//    4 waves / block; block owns 64 M-rows.
//    K staged in double-buffered LDS via GLOBAL_LOAD_ASYNC_TO_LDS_B128:
//    next chunk's copies are in flight while WMMAs consume the current one.
//    A frag (16x4 f32): lanes 0-15 hold K=k,k+1 ; lanes 16-31 hold K=k+2,k+3.
//    C frag: VGPR j <-> M=j (lanes 0-15) / M=8+j (lanes 16-31), N = lane%16.
// ---------------------------------------------------------------------------
__global__ void __launch_bounds__(128)
sampled_gemm_kernel(const float* __restrict__ inputs,
                    const float* __restrict__ sw,
                    const float* __restrict__ colbias,
                    float* __restrict__ out) {
  __shared__ float lA[2][64 * LDT];      // 2 x 9.2 KB
  __shared__ float lB[2][SPAD * LDT];    // 2 x 16.1 KB  (50.7 KB total)

  const int tid   = threadIdx.x;
  const int wave  = tid >> 5;
  const int lane  = tid & 31;
  const int half  = lane >> 4;
  const int l16   = lane & 15;
  const int mbase = blockIdx.x * 64;

  // Issue all async b128 copies for one K-chunk into buffer `buf`.
  auto stage = [&](int buf, int kc) {
    // A tile: 64 rows x KC cols = 512 x b128, 4 per thread
#pragma unroll
    for (int i = 0; i < 4; ++i) {
      int f  = tid + 128 * i;          // [0, 512)
      int r  = f >> 3;                 // 8 float4 per row
      int c4 = f & 7;
      async_ld128((uint32_t)(uintptr_t)&lA[buf][r * LDT + c4 * 4],
                  inputs + (size_t)(mbase + r) * DIM + kc + c4 * 4);
    }
    // B tile: 112 rows x KC cols = 896 x b128, 7 per thread
#pragma unroll
    for (int i = 0; i < 7; ++i) {
      int f  = tid + 128 * i;          // [0, 896)
      int r  = f >> 3;
      int c4 = f & 7;
      async_ld128((uint32_t)(uintptr_t)&lB[buf][r * LDT + c4 * 4],
                  sw + (size_t)r * DIM + kc + c4 * 4);
    }
  };

  v8f acc[NTILES] = {};

  stage(0, 0);
  wait_async0();
  __syncthreads();

  for (int kc = 0; kc < DIM; kc += KC) {
    const int buf = (kc / KC) & 1;
    // Prefetch next chunk into the other buffer (safe: the barrier that ended
    // the previous iteration guarantees nobody still reads buf^1).
    if (kc + KC < DIM) stage(buf ^ 1, kc + KC);

    const float* aRow  = &lA[buf][(wave * 16 + l16) * LDT];
    const float* bBase = &lB[buf][0];
#pragma unroll
    for (int ks = 0; ks < KC; ks += 4) {
      v2f a = *(const v2f*)(aRow + ks + 2 * half);     // 8B-aligned ds_load
#pragma unroll
      for (int t = 0; t < NTILES; ++t) {
        v2f b = *(const v2f*)(bBase + (t * 16 + l16) * LDT + ks + 2 * half);
        acc[t] = __builtin_amdgcn_wmma_f32_16x16x4_f32(
            /*neg_a=*/false, a, /*neg_b=*/false, b,
            /*c_mod=*/(short)0, acc[t], /*reuse_a=*/false, /*reuse_b=*/false);
      }
    }
    wait_async0();       // my in-flight copies for the next chunk are done
    __syncthreads();     // everyone's copies done, everyone done reading buf
  }

  // write back: out[m][1 + s] = acc + (bias_s - log q_s)
#pragma unroll
  for (int t = 0; t < NTILES; ++t) {
    int s = t * 16 + l16;
    if (s < NSAMP) {
      float cb = colbias[s];
#pragma unroll
      for (int j = 0; j < 8; ++j) {
        int m = mbase + wave * 16 + half * 8 + j;
        out[(size_t)m * COLS + 1 + s] = acc[t][j] + cb;
      }
    }
  }
}

// ---------------------------------------------------------------------------
// 4) Per-row logsumexp - true_logit; 8 rows per 256-thread block;
//    deterministic partial per block.
// ---------------------------------------------------------------------------
__global__ void loss_partial_kernel(const float* __restrict__ logits,
                                    float* __restrict__ partials) {
  __shared__ float sh[8];
  int wib  = threadIdx.x >> 5;
  int lane = threadIdx.x & 31;
  int wid  = blockIdx.x * 8 + wib;       // < B_ROWS by construction
  const float* row = logits + (size_t)wid * COLS;

  float v[4];
  float m = -__builtin_inff();
#pragma unroll
  for (int i = 0; i < 4; ++i) {
    int c = lane + 32 * i;
    v[i] = (c < COLS) ? row[c] : -__builtin_inff();
    m = fmaxf(m, v[i]);
  }
#pragma unroll
  for (int off = 16; off > 0; off >>= 1) m = fmaxf(m, __shfl_xor(m, off, 32));
  float e = 0.f;
#pragma unroll
  for (int i = 0; i < 4; ++i) e += expf(v[i] - m);   // exp(-inf)=0 for pads
#pragma unroll
  for (int off = 16; off > 0; off >>= 1) e += __shfl_xor(e, off, 32);
  float loss = m + logf(e) - row[0];

  if (lane == 0) sh[wib] = loss;
  __syncthreads();
  if (threadIdx.x == 0) {
    float t = 0.f;
#pragma unroll
    for (int i = 0; i < 8; ++i) t += sh[i];
    partials[blockIdx.x] = t;
  }
}

// ---------------------------------------------------------------------------
// 5) Final deterministic reduction -> mean loss.
// ---------------------------------------------------------------------------
__global__ void loss_final_kernel(const float* __restrict__ partials,
                                  float* __restrict__ loss_out) {
  __shared__ float sh[256];
  float s = 0.f;
  for (int i = threadIdx.x; i < B_ROWS / 8; i += 256) s += partials[i];
  sh[threadIdx.x] = s;
  __syncthreads();
  for (int off = 128; off > 0; off >>= 1) {
    if (threadIdx.x < (unsigned)off) sh[threadIdx.x] += sh[threadIdx.x + off];
    __syncthreads();
  }
  if (threadIdx.x == 0) loss_out[0] = sh[0] / (float)B_ROWS;
}

// ---------------------------------------------------------------------------
extern "C" void kernel_launch(void* const* d_in, const int* in_sizes, int n_in,
                              void* d_out, int out_size, void* d_ws, size_t ws_size,
                              hipStream_t stream) {
  const int*   label       = (const int*)d_in[0];
  const float* inputs      = (const float*)d_in[1];
  const float* weights     = (const float*)d_in[2];
  const float* biases      = (const float*)d_in[3];
  const int*   sampled_ids = (const int*)d_in[4];
  const float* texp        = (const float*)d_in[5];
  const float* sexp        = (const float*)d_in[6];
  float* out = (float*)d_out;

  float* ws       = (float*)d_ws;
  float* sw       = ws;                        // SPAD*DIM floats
  float* colbias  = sw + SPAD * DIM;           // SPAD floats
  float* partials = colbias + SPAD;            // B_ROWS/8 floats

  gather_kernel<<<(SPAD * DIM) / 256, 256, 0, stream>>>(
      weights, biases, sampled_ids, sexp, sw, colbias);

  true_logits_kernel<<<(B_ROWS * 32) / 256, 256, 0, stream>>>(
      inputs, weights, biases, label, texp, out);

  sampled_gemm_kernel<<<B_ROWS / 64, 128, 0, stream>>>(
      inputs, sw, colbias, out);

  loss_partial_kernel<<<B_ROWS / 8, 256, 0, stream>>>(out, partials);

  loss_final_kernel<<<1, 256, 0, stream>>>(partials, out + (size_t)B_ROWS * COLS);
}